// BallQueryGrouper_9242769621829
// MI455X (gfx1250) — compile-verified
//
#include <hip/hip_runtime.h>
#include <stdint.h>

// Problem constants (from the reference setup)
#define B_   4
#define M_   2048
#define N_   8192
#define C_   64
#define S_   32        // NSAMPLE
#define R2_  0.01f     // RADIUS^2
#define OUTW 67        // 3 + C

typedef int v4i __attribute__((vector_size(16)));

// ---- CDNA5 async global->LDS copy (ASYNCcnt-tracked) ------------------------
__device__ __forceinline__ void async_load_b128_to_lds(const void* gsrc, void* ldst) {
#if __has_builtin(__builtin_amdgcn_global_load_async_to_lds_b128)
  __builtin_amdgcn_global_load_async_to_lds_b128(
      (v4i*)gsrc, (v4i*)ldst, /*imm offset*/0, /*cpol*/0);
#else
  unsigned loff = (unsigned)(unsigned long long)ldst;
  asm volatile("global_load_async_to_lds_b128 %0, %1, off"
               :: "v"(loff), "v"(gsrc) : "memory");
#endif
}

__device__ __forceinline__ void wait_asynccnt_zero() {
#if __has_builtin(__builtin_amdgcn_s_wait_asynccnt)
  __builtin_amdgcn_s_wait_asynccnt(0);
#else
  asm volatile("s_wait_asynccnt 0" ::: "memory");
#endif
}

// -----------------------------------------------------------------------------
// One block = 1 batch slice, 32 query points (4 waves x 8 queries).
// LDS holds the full per-batch pointset (8192 x 3 f32 = 96 KB; fits in CDNA5's
// 320 KB WGP LDS, 3 blocks/WGP occupancy).
// -----------------------------------------------------------------------------
__global__ __launch_bounds__(128)
void ball_query_group_kernel(const float* __restrict__ new_xyz,   // [B,M,3]
                             const float* __restrict__ pointset,  // [B,N,3]
                             const float* __restrict__ feature,   // [B,N,C]
                             float* __restrict__ out) {           // [B,M,S,67]
  __shared__ float sxyz[N_ * 3];   // 96 KB, interleaved xyz (bank-conflict-free)

  const int tid   = threadIdx.x;
  const int lane  = tid & 31;
  const int wave  = tid >> 5;
  const int b     = blockIdx.x >> 6;           // 64 blocks per batch
  const int mbase = (blockIdx.x & 63) * 32;    // 32 queries per block

  // ---- Stage pointset[b] (96 KB) into LDS: exactly 48 async b128s/thread ----
  {
    const char* src = (const char*)(pointset + (size_t)b * (N_ * 3)) + (size_t)tid * 16;
    char*       dst = (char*)sxyz + (size_t)tid * 16;
    // 6144 x 16B chunks / 128 threads = 48 per thread; constant offsets fold
    // into each instruction's IOFFSET (single burst, ASYNCcnt <= 48 < 63).
#pragma unroll
    for (int it = 0; it < 48; ++it) {
      async_load_b128_to_lds(src + (size_t)it * (128 * 16),
                             dst + (size_t)it * (128 * 16));
    }
    wait_asynccnt_zero();
  }
  __syncthreads();

  const float* featB = feature + (size_t)b * N_ * C_;

  // ---- Each wave processes 8 query points ----
  for (int qi = 0; qi < 8; ++qi) {
    const int m = mbase + wave * 8 + qi;
    const float* q = new_xyz + ((size_t)b * M_ + m) * 3;
    const float qx = q[0], qy = q[1], qz = q[2];

    // Ball query: collect first S_ indices (ascending) with d2 < R2.
    // Lane `l` owns sample slot `l`.
    int my_idx = 0;
    int cnt    = 0;
    for (int base = 0; base < N_; base += 32) {
      const int j = base + lane;
      const float dx = sxyz[j * 3 + 0] - qx;
      const float dy = sxyz[j * 3 + 1] - qy;
      const float dz = sxyz[j * 3 + 2] - qz;
      const bool hit = (dx * dx + dy * dy + dz * dz) < R2_;

      const unsigned mask = (unsigned)__ballot(hit);  // wave32: low 32 bits
      const int nh = __popc(mask);
      const int r  = lane - cnt;            // hit-rank this slot needs
      if (r >= 0 && r < nh) {
        unsigned mm = mask;
        for (int t = 0; t < r; ++t) mm &= (mm - 1u);  // drop r lowest set bits
        my_idx = base + (__ffs(mm) - 1);              // (r+1)-th hit lane
      }
      cnt += nh;
      if (cnt >= S_) break;                 // uniform early exit
    }
    if (cnt > S_) cnt = S_;

    // Back-fill unfilled slots with the first found index (0 if none found).
    const int idx0 = __shfl(my_idx, 0);
    if (lane >= cnt) my_idx = idx0;

    // ---- Gather + write: 32 samples x 67 floats, 3 coalesced passes each ----
    const float qc = (lane == 0) ? qx : ((lane == 1) ? qy : qz);
    float* orow = out + ((size_t)b * M_ + m) * (size_t)(S_ * OUTW);
    for (int s = 0; s < S_; ++s) {
      const int k = __shfl(my_idx, s);
      const float* frow = featB + (size_t)k * C_;
      float* op = orow + (size_t)s * OUTW;

      // t = lane (0..31): t<3 -> local xyz, else feat[t-3]
      const int li = (lane < 3) ? lane : 0;        // clamp (avoid spec. OOB)
      const int fi = (lane < 3) ? 0 : (lane - 3);
      const float v0 = (lane < 3) ? (sxyz[k * 3 + li] - qc) : frow[fi];
      __builtin_nontemporal_store(v0, op + lane);

      // t = lane + 32 (32..63): feat[t-3] = feat[lane+29]
      const float v1 = frow[lane + 29];
      __builtin_nontemporal_store(v1, op + lane + 32);

      // t = lane + 64 (64..66): lanes 0..2 -> feat[lane+61]
      if (lane < 3) {
        __builtin_nontemporal_store(frow[lane + 61], op + lane + 64);
      }
    }
  }
}

extern "C" void kernel_launch(void* const* d_in, const int* in_sizes, int n_in,
                              void* d_out, int out_size, void* d_ws, size_t ws_size,
                              hipStream_t stream) {
  (void)in_sizes; (void)n_in; (void)out_size; (void)d_ws; (void)ws_size;
  const float* new_xyz  = (const float*)d_in[0];  // [B,M,3]
  const float* pointset = (const float*)d_in[1];  // [B,N,3]
  const float* feature  = (const float*)d_in[2];  // [B,N,C]
  float* out = (float*)d_out;                     // [B,M,S,67]

  dim3 grid(B_ * (M_ / 32));   // 256 blocks
  dim3 block(128);             // 4 waves (wave32)
  hipLaunchKernelGGL(ball_query_group_kernel, grid, block, 0, stream,
                     new_xyz, pointset, feature, out);
}